// BeamSearch_4363686772853
// MI455X (gfx1250) — compile-verified
//
#include <hip/hip_runtime.h>
#include <stdint.h>

#define T 256
#define V 5000
#define BEAM 16
#define W_ASR 0.7f
#define W_LM 0.3f
#define SOS 1
#define NEG_INF (-1e30f)

#define NTH 512          // main kernel threads (16 waves, wave32)
#define NWAVES (NTH / 32)
#define VCHUNK 10        // ceil(5000/512) async-load slices per asr row

// ---------- helpers ----------

__device__ __forceinline__ unsigned shfl_xor_u32(unsigned x, int m) {
  return (unsigned)__shfl_xor((int)x, m, 32);
}
__device__ __forceinline__ unsigned long long shfl_xor_u64(unsigned long long x, int m) {
  unsigned lo = (unsigned)x, hi = (unsigned)(x >> 32);
  lo = shfl_xor_u32(lo, m);
  hi = shfl_xor_u32(hi, m);
  return ((unsigned long long)hi << 32) | (unsigned long long)lo;
}
__device__ __forceinline__ unsigned long long wave_max_u64(unsigned long long k) {
#pragma unroll
  for (int off = 16; off > 0; off >>= 1) {
    unsigned long long o = shfl_xor_u64(k, off);
    k = (o > k) ? o : k;
  }
  return k;
}
__device__ __forceinline__ float wave_max_f(float v) {
#pragma unroll
  for (int off = 16; off > 0; off >>= 1) v = fmaxf(v, __shfl_xor(v, off, 32));
  return v;
}
__device__ __forceinline__ float wave_sum_f(float v) {
#pragma unroll
  for (int off = 16; off > 0; off >>= 1) v += __shfl_xor(v, off, 32);
  return v;
}

// monotone float->u32 mapping (order-preserving over all finite floats)
__device__ __forceinline__ unsigned ord_enc(float f) {
  unsigned b = __float_as_uint(f);
  return (b & 0x80000000u) ? ~b : (b | 0x80000000u);
}
__device__ __forceinline__ float ord_dec(unsigned e) {
  unsigned b = (e & 0x80000000u) ? (e ^ 0x80000000u) : ~e;
  return __uint_as_float(b);
}

// ---------- pass 1: per-row logsumexp (for both asr [256 x 5000] and lm [5000 x 5000]) ----------

__global__ __launch_bounds__(256)
void lse_rows_kernel(const float* __restrict__ mat, float* __restrict__ out) {
  const int row = blockIdx.x;
  const int tid = threadIdx.x;
  const float* __restrict__ r = mat + (size_t)row * V;
  __shared__ float s_red[8];

  float m = -INFINITY;
  for (int i = tid; i < V; i += 256) m = fmaxf(m, r[i]);
  m = wave_max_f(m);
  if ((tid & 31) == 0) s_red[tid >> 5] = m;
  __syncthreads();
  float mm = s_red[0];
#pragma unroll
  for (int w = 1; w < 8; ++w) mm = fmaxf(mm, s_red[w]);
  __syncthreads();  // s_red reuse

  float s = 0.f;
  for (int i = tid; i < V; i += 256) s += expf(r[i] - mm);
  s = wave_sum_f(s);
  if ((tid & 31) == 0) s_red[tid >> 5] = s;
  __syncthreads();
  if (tid == 0) {
    float tot = 0.f;
#pragma unroll
    for (int w = 0; w < 8; ++w) tot += s_red[w];
    out[row] = mm + logf(tot);
  }
}

// ---------- pass 2: persistent single-workgroup beam search ----------

__global__ __launch_bounds__(NTH)
void beam_kernel(const float* __restrict__ asr, const float* __restrict__ lm,
                 const float* __restrict__ lse_asr, const float* __restrict__ lse_lm,
                 float* __restrict__ out) {
  __shared__ float s_asr[2][V];                 // double-buffered asr row (40 KB)
  __shared__ unsigned long long s_red[NWAVES];  // cross-wave reduce
  __shared__ unsigned long long s_win[BEAM];    // sorted step winners
  __shared__ float s_scores[BEAM];
  __shared__ int s_last[BEAM];
  __shared__ float s_base[BEAM];
  __shared__ unsigned s_rowoff[BEAM];
  __shared__ unsigned short s_tok[T * BEAM];    // 8 KB token history
  __shared__ unsigned char s_bp[T * BEAM];      // 4 KB backpointer history

  const int tid = threadIdx.x;
  const int lane = tid & 31;
  const int wave = tid >> 5;

  if (tid < BEAM) {
    s_scores[tid] = (tid == 0) ? 0.f : NEG_INF;
    s_last[tid] = SOS;
  }

  // LDS byte offsets of the two asr staging buffers (low 32 bits of generic ptr)
  const unsigned lds0 = (unsigned)(size_t)(&s_asr[0][0]);
  const unsigned lds1 = (unsigned)(size_t)(&s_asr[1][0]);

  // prologue: async-stage asr row 0 into buffer 0 (exactly VCHUNK issues per lane)
  {
    const float* g = asr;
#pragma unroll
    for (int j = 0; j < VCHUNK; ++j) {
      int idx = tid + j * NTH;
      idx = (idx < V) ? idx : (V - 1);  // clamped duplicates keep ASYNCcnt uniform
      unsigned l = lds0 + (unsigned)idx * 4u;
      const float* gp = g + idx;
      asm volatile("global_load_async_to_lds_b32 %0, %1, off"
                   :: "v"(l), "v"(gp) : "memory");
    }
  }
  __syncthreads();

  for (int t = 0; t < T; ++t) {
    // issue async stage of row t+1 into the other buffer (its previous contents
    // were last read in step t-1, fenced by the end-of-step barrier)
    {
      const int nrow = (t + 1 < T) ? (t + 1) : (T - 1);
      const unsigned ldsn = ((t + 1) & 1) ? lds1 : lds0;
      const float* g = asr + (size_t)nrow * V;
#pragma unroll
      for (int j = 0; j < VCHUNK; ++j) {
        int idx = tid + j * NTH;
        idx = (idx < V) ? idx : (V - 1);
        unsigned l = ldsn + (unsigned)idx * 4u;
        const float* gp = g + idx;
        asm volatile("global_load_async_to_lds_b32 %0, %1, off"
                     :: "v"(l), "v"(gp) : "memory");
      }
    }

    // per-beam merged base (same thread wrote s_scores/s_last last step: no sync needed)
    if (tid < BEAM) {
      const int lastb = s_last[tid];
      s_base[tid] = s_scores[tid] - W_ASR * lse_asr[t] - W_LM * lse_lm[lastb];
      s_rowoff[tid] = (unsigned)lastb * (unsigned)V;
    }

    // wait for row t's stage: VCHUNK newer (row t+1) issues may remain outstanding
    asm volatile("s_wait_asynccnt 0xA" ::: "memory");
    __syncthreads();

    // warm the 16 gathered LM rows toward the WGP (L2-resident after pass 1)
#pragma unroll 1
    for (int b = 0; b < BEAM; ++b) {
      const char* rp = (const char*)(lm + (size_t)s_rowoff[b]);
      for (int off = tid * 128; off < V * 4; off += NTH * 128)
        __builtin_prefetch(rp + off, 0, 3);
    }

    // candidate evaluation + thread-local sorted top-16 of packed keys
    unsigned long long loc[BEAM];
#pragma unroll
    for (int i = 0; i < BEAM; ++i) loc[i] = 0ull;

    const float* __restrict__ abuf = &s_asr[t & 1][0];
    for (int j = 0; j < VCHUNK; ++j) {
      const int v = tid + j * NTH;
      if (v >= V) break;
      const float a = W_ASR * abuf[v];
#pragma unroll
      for (int b = 0; b < BEAM; ++b) {
        const float c = s_base[b] + a + W_LM * lm[(size_t)s_rowoff[b] + v];
        const unsigned flat = (unsigned)b * (unsigned)V + (unsigned)v;
        const unsigned long long k =
            ((unsigned long long)ord_enc(c) << 32) |
            (unsigned long long)(0xFFFFFFFFu - flat);  // smaller flat idx wins ties
        if (k > loc[BEAM - 1]) {
          int p = BEAM - 1;
          while (p > 0 && loc[p - 1] < k) { loc[p] = loc[p - 1]; --p; }
          loc[p] = k;
        }
      }
    }

    // 16-round k-way merge of the 512 sorted local lists -> global sorted top-16
    int h = 0;
    for (int r = 0; r < BEAM; ++r) {
      const unsigned long long mine = (h < BEAM) ? loc[h] : 0ull;
      unsigned long long k = wave_max_u64(mine);
      if (lane == 0) s_red[wave] = k;
      __syncthreads();
      if (wave == 0) {
        unsigned long long kk = (lane < NWAVES) ? s_red[lane] : 0ull;
        kk = wave_max_u64(kk);
        if (lane == 0) s_win[r] = kk;
      }
      __syncthreads();
      if (mine != 0ull && mine == s_win[r]) ++h;  // unique owner advances
    }

    // decode winners -> new state + history
    if (tid < BEAM) {
      const unsigned long long k = s_win[tid];
      const unsigned flat = 0xFFFFFFFFu - (unsigned)(k & 0xFFFFFFFFu);
      const int b = (int)(flat / (unsigned)V);
      const int v2 = (int)(flat % (unsigned)V);
      s_scores[tid] = ord_dec((unsigned)(k >> 32));
      s_last[tid] = v2;
      s_tok[t * BEAM + tid] = (unsigned short)v2;
      s_bp[t * BEAM + tid] = (unsigned char)b;
    }
    __syncthreads();  // end-of-step fence (also protects asr buffer reuse)
  }

  // backtrack + emit: out[0..15] = sorted nbest scores; out[16 + b*T + t] = token
  if (tid < BEAM) {
    out[tid] = s_scores[tid];
    int ptr = tid;
    for (int t2 = T - 1; t2 >= 0; --t2) {
      out[BEAM + tid * T + t2] = (float)s_tok[t2 * BEAM + ptr];
      ptr = (int)s_bp[t2 * BEAM + ptr];
    }
  }
}

// ---------- launch ----------

extern "C" void kernel_launch(void* const* d_in, const int* in_sizes, int n_in,
                              void* d_out, int out_size, void* d_ws, size_t ws_size,
                              hipStream_t stream) {
  const float* asr = (const float*)d_in[0];  // [256, 5000]
  const float* lm = (const float*)d_in[1];   // [5000, 5000]
  float* ws = (float*)d_ws;
  float* lse_lm = ws;            // 5000 floats
  float* lse_asr = ws + V;       // 256 floats

  lse_rows_kernel<<<V, 256, 0, stream>>>(lm, lse_lm);
  lse_rows_kernel<<<T, 256, 0, stream>>>(asr, lse_asr);
  beam_kernel<<<1, NTH, 0, stream>>>(asr, lm, lse_asr, lse_lm, (float*)d_out);
}